// ReL_15547781612172
// MI455X (gfx1250) — compile-verified
//
#include <hip/hip_runtime.h>

// CDNA5 / gfx1250: wave32, WMMA 16x16x4 f32.
typedef __attribute__((ext_vector_type(2))) float v2f;
typedef __attribute__((ext_vector_type(8))) float v8f;

#define NBLK 128
#define NTHR 256
#define NWAVE (NTHR / 32)

// Stage 1: per-thread gather proba[i, target[i]] * reward[i], then a wave32
// reduction whose first step is V_WMMA_F32_16X16X4_F32 with A = ones:
// D[m][n] = sum_k B[k][n], so summing D row 0 over n yields the sum of all
// 64 B slots (= the wave's 2x32 partial accumulators), in exact f32 FMA math.
__global__ __launch_bounds__(NTHR) void gather_partial_kernel(
    const float* __restrict__ proba,
    const float* __restrict__ reward,
    const int* __restrict__ target,
    float* __restrict__ partials,
    long long length, long long vocab)
{
    const long long stride = (long long)gridDim.x * (long long)blockDim.x;
    long long i = (long long)blockIdx.x * blockDim.x + threadIdx.x;

    float s0 = 0.0f, s1 = 0.0f;
    // Unroll-by-2 grid-stride loop feeding two accumulators (both WMMA B slots).
    while (i + stride < length) {
        {
            const long long t = (long long)target[i];
            const float p = __builtin_nontemporal_load(&proba[i * vocab + t]);
            s0 = fmaf(p, reward[i], s0);
        }
        {
            const long long j = i + stride;
            const long long t = (long long)target[j];
            const float p = __builtin_nontemporal_load(&proba[j * vocab + t]);
            s1 = fmaf(p, reward[j], s1);
        }
        i += 2 * stride;
    }
    if (i < length) {
        const long long t = (long long)target[i];
        const float p = __builtin_nontemporal_load(&proba[i * vocab + t]);
        s0 = fmaf(p, reward[i], s0);
    }

    // ---- wave32 reduction, step 1: one WMMA folds 64 partials -> 16 colsums.
    // EXEC is all-ones here (no divergence at this point), as WMMA requires.
    v2f a; a.x = 1.0f; a.y = 1.0f;   // A = 16x4 ones
    v2f b; b.x = s0;  b.y = s1;      // B = 4x16, every slot covered once
    v8f c = {};
    c = __builtin_amdgcn_wmma_f32_16x16x4_f32(
        /*neg_a=*/false, a, /*neg_b=*/false, b,
        /*c_mod=*/(short)0, c, /*reuse_a=*/false, /*reuse_b=*/false);

    // D row M=0 lives in c[0], lanes 0-15 (lanes 16-31 hold row M=8, which is
    // identical since every A row is ones). 4 xor-shuffles finish 16 -> 1.
    float r = c[0];
    r += __shfl_xor(r, 1, 32);
    r += __shfl_xor(r, 2, 32);
    r += __shfl_xor(r, 4, 32);
    r += __shfl_xor(r, 8, 32);

    // ---- block reduction over 8 waves via LDS.
    __shared__ float wsum[NWAVE];
    const int lane = threadIdx.x & 31;
    const int wv   = threadIdx.x >> 5;
    if (lane == 0) wsum[wv] = r;
    __syncthreads();
    if (threadIdx.x == 0) {
        float s = 0.0f;
#pragma unroll
        for (int k = 0; k < NWAVE; ++k) s += wsum[k];
        partials[blockIdx.x] = s;
    }
}

// Stage 2: deterministic final reduction of NBLK partials; writes -sum.
__global__ __launch_bounds__(128) void finalize_kernel(
    const float* __restrict__ partials, int n, float* __restrict__ out)
{
    float v = 0.0f;
    for (int i = threadIdx.x; i < n; i += blockDim.x) v += partials[i];
    v += __shfl_xor(v, 1, 32);
    v += __shfl_xor(v, 2, 32);
    v += __shfl_xor(v, 4, 32);
    v += __shfl_xor(v, 8, 32);
    v += __shfl_xor(v, 16, 32);

    __shared__ float wsum[4];
    const int lane = threadIdx.x & 31;
    const int wv   = threadIdx.x >> 5;
    if (lane == 0) wsum[wv] = v;
    __syncthreads();
    if (threadIdx.x == 0) {
        float s = 0.0f;
#pragma unroll
        for (int k = 0; k < 4; ++k) s += wsum[k];
        *out = -s;
    }
}

extern "C" void kernel_launch(void* const* d_in, const int* in_sizes, int n_in,
                              void* d_out, int out_size, void* d_ws, size_t ws_size,
                              hipStream_t stream) {
    const float* proba  = (const float*)d_in[0];  // [length, vocab] f32
    const float* reward = (const float*)d_in[1];  // [length] f32
    const int*   target = (const int*)d_in[2];    // [length] int (harness int cast)
    // d_in[3] = vocab_size scalar on device; derive shapes host-side instead.
    const long long length = (long long)in_sizes[1];
    const long long vocab  = length > 0 ? (long long)in_sizes[0] / length : 0;

    float* partials = (float*)d_ws;               // NBLK floats of scratch
    float* out      = (float*)d_out;

    gather_partial_kernel<<<NBLK, NTHR, 0, stream>>>(
        proba, reward, target, partials, length, vocab);
    finalize_kernel<<<1, 128, 0, stream>>>(partials, NBLK, out);
}